// Model_7430293422621
// MI455X (gfx1250) — compile-verified
//
#include <hip/hip_runtime.h>
#include <math.h>

// ---------------------------------------------------------------------------
// Problem constants (from reference): B=8, H=W=128, C1=16, C2=64, C3=32,
// pool 16 -> 8x8, feat_dim = 32*8*8 = 2048, 2 classes.
// ---------------------------------------------------------------------------
#define BATCH   8
#define HDIM    128
#define WDIM    128
#define HW      16384           // 128*128
#define NPIX    131072          // 8*128*128
#define EPSBN   1e-5f

typedef float v2f __attribute__((ext_vector_type(2)));
typedef float v8f __attribute__((ext_vector_type(8)));

// Workspace layout (float offsets)
#define OFF_C1    ((size_t)0)                       // 8*16*HW  = 2,097,152
#define OFF_C2    (OFF_C1 + (size_t)2097152)        // 8*64*HW  = 8,388,608
#define OFF_FEAT  (OFF_C2 + (size_t)8388608)        // 8*32*HW  = 4,194,304
#define OFF_MU1   (OFF_FEAT + (size_t)4194304)
#define OFF_ISTD1 (OFF_MU1 + 16)
#define OFF_MU2   (OFF_ISTD1 + 16)
#define OFF_ISTD2 (OFF_MU2 + 64)
#define OFF_MU3   (OFF_ISTD2 + 64)
#define OFF_ISTD3 (OFF_MU3 + 32)
#define OFF_P     (OFF_ISTD3 + 32)                  // 8*2048 = 16384
#define OFF_GF    (OFF_P + 16384)                   // 8*2*2048 = 32768
#define OFF_S1    (OFF_GF + 32768)                  // 8*2*32 = 512
#define OFF_S2    (OFF_S1 + 512)                    // 512

// ---------------------------------------------------------------------------
// conv1: 3->16, 2x2 SAME (pad high). Direct VALU, K=12 is too small for WMMA.
// One thread per output pixel, all 16 out channels.
// ---------------------------------------------------------------------------
__global__ __launch_bounds__(256) void conv1_kernel(
    const float* __restrict__ x, const float* __restrict__ w1,
    const float* __restrict__ b1, float* __restrict__ c1)
{
    __shared__ float sw[192];            // w1 (16,3,2,2)
    const int tid = threadIdx.x;
    if (tid < 192) sw[tid] = w1[tid];
    __syncthreads();

    const int p  = blockIdx.x * 256 + tid;     // 0..NPIX-1
    const int n  = p >> 14;
    const int hw = p & (HW - 1);
    const int h  = hw >> 7;
    const int w  = hw & 127;

    float in[3][4];
    #pragma unroll
    for (int c = 0; c < 3; ++c) {
        #pragma unroll
        for (int kp = 0; kp < 4; ++kp) {
            const int hh = h + (kp >> 1), ww = w + (kp & 1);
            in[c][kp] = (hh < HDIM && ww < WDIM)
                ? x[((size_t)(n * 3 + c)) * HW + hh * WDIM + ww] : 0.0f;
        }
    }
    #pragma unroll
    for (int o = 0; o < 16; ++o) {
        float acc = b1[o];
        #pragma unroll
        for (int c = 0; c < 3; ++c)
            #pragma unroll
            for (int kp = 0; kp < 4; ++kp)
                acc = fmaf(sw[o * 12 + c * 4 + kp], in[c][kp], acc);
        c1[((size_t)(n * 16 + o)) * HW + hw] = acc;
    }
}

// ---------------------------------------------------------------------------
// Per-channel batch stats (training-mode BN, biased variance).
// One block per channel; reduce sum & sumsq over B*H*W.
// ---------------------------------------------------------------------------
__global__ __launch_bounds__(256) void bn_stats_kernel(
    const float* __restrict__ src, int C,
    float* __restrict__ mu, float* __restrict__ istd)
{
    const int c = blockIdx.x, tid = threadIdx.x;
    float s = 0.0f, s2 = 0.0f;
    for (int nn = 0; nn < BATCH; ++nn) {
        const float* base = src + ((size_t)(nn * C + c)) * HW;
        for (int p = tid; p < HW; p += 256) {
            const float v = base[p];
            s += v; s2 += v * v;
        }
    }
    __shared__ float rs[256], rs2[256];
    rs[tid] = s; rs2[tid] = s2;
    __syncthreads();
    for (int o = 128; o > 0; o >>= 1) {
        if (tid < o) { rs[tid] += rs[tid + o]; rs2[tid] += rs2[tid + o]; }
        __syncthreads();
    }
    if (tid == 0) {
        const float N = (float)(BATCH * HW);
        const float m = rs[0] / N;
        const float var = rs2[0] / N - m * m;
        mu[c] = m;
        istd[c] = rsqrtf(var + EPSBN);
    }
}

// ---------------------------------------------------------------------------
// conv2 via fp32 WMMA: (16 pixels) x (16 outch) tiles, K = 16ch*4taps = 64.
// Block = 128 threads = 4 waves covering out-channels 0..63 of one pixel tile.
// BN1 is folded as affine while staging the input tile into LDS.
// WMMA fragment layouts per CDNA5 ISA 7.12.2 (32-bit A 16x4, B 4x16, C/D 8 VGPR).
// ---------------------------------------------------------------------------
__global__ __launch_bounds__(128) void conv2_wmma_kernel(
    const float* __restrict__ c1, const float* __restrict__ w2,
    const float* __restrict__ b2, const float* __restrict__ g1,
    const float* __restrict__ bt1, const float* __restrict__ mu1,
    const float* __restrict__ istd1, float* __restrict__ c2)
{
    __shared__ float sIn[16][2][18];   // [cin][ki][col 0..16]
    __shared__ float sW[64][64];       // [kk = c*4+ki*2+kj][o]

    const int T  = blockIdx.x;         // 8192 pixel tiles
    const int n  = T >> 10;
    const int h  = (T >> 3) & 127;
    const int j0 = (T & 7) << 4;
    const int tid = threadIdx.x;

    // weights: w2 is (64,16,2,2) -> flat o*64 + kk
    for (int idx = tid; idx < 64 * 64; idx += 128) {
        const int o = idx >> 6, kk = idx & 63;
        sW[kk][o] = w2[o * 64 + kk];
    }
    // input tile: rows h,h+1, cols j0..j0+16 (SAME pad high -> zeros), BN1 affine
    for (int idx = tid; idx < 16 * 34; idx += 128) {
        const int c = idx / 34, r = idx % 34, rr = r / 17, t = r % 17;
        const int hh = h + rr, ww = j0 + t;
        float v = 0.0f;
        if (hh < HDIM && ww < WDIM)
            v = c1[((size_t)(n * 16 + c)) * HW + hh * WDIM + ww];
        const float a = g1[c] * istd1[c];
        sIn[c][rr][t] = fmaf(a, v, bt1[c] - mu1[c] * a);
    }
    __syncthreads();

    const int wave = tid >> 5, lane = tid & 31;
    const int m = lane & 15, hi = lane >> 4;     // hi selects K-half per ISA layout
    const int o0 = wave << 4;

    v8f acc = {0.f, 0.f, 0.f, 0.f, 0.f, 0.f, 0.f, 0.f};
    #pragma unroll 4
    for (int c = 0; c < 16; ++c) {
        v2f a, b;
        a.x = sIn[c][hi][m];          // A[m][c*4 + hi*2 + 0]
        a.y = sIn[c][hi][m + 1];      // A[m][c*4 + hi*2 + 1]
        const int kk = (c << 2) + (hi << 1);
        b.x = sW[kk][o0 + m];         // B[kk][N]
        b.y = sW[kk + 1][o0 + m];
        acc = __builtin_amdgcn_wmma_f32_16x16x4_f32(
            false, a, false, b, (short)0, acc, false, false);
    }
    // D layout: lane(hi,m): N = o0+m, M = hi*8 + r  (consecutive pixels in w)
    const float bias = b2[o0 + m];
    float* dst = c2 + ((size_t)(n * 64 + o0 + m)) * HW + h * WDIM + j0 + (hi << 3);
    #pragma unroll
    for (int r = 0; r < 8; ++r) dst[r] = acc[r] + bias;
}

// ---------------------------------------------------------------------------
// conv3 via fp32 WMMA: K = 64ch*4taps = 256, Cout=32 -> 2 waves per block.
// Produces the hooked feature tensor (pre-BN3, bias included).
// ---------------------------------------------------------------------------
__global__ __launch_bounds__(64) void conv3_wmma_kernel(
    const float* __restrict__ c2, const float* __restrict__ w3,
    const float* __restrict__ b3, const float* __restrict__ g2,
    const float* __restrict__ bt2, const float* __restrict__ mu2,
    const float* __restrict__ istd2, float* __restrict__ feat)
{
    __shared__ float sIn[64][2][18];   // 9.2 KB
    __shared__ float sW[256][32];      // 32 KB

    const int T  = blockIdx.x;
    const int n  = T >> 10;
    const int h  = (T >> 3) & 127;
    const int j0 = (T & 7) << 4;
    const int tid = threadIdx.x;       // 0..63

    for (int idx = tid; idx < 256 * 32; idx += 64) {
        const int o = idx & 31, kk = idx >> 5;
        sW[kk][o] = w3[o * 256 + kk];  // w3 (32,64,2,2) -> o*256 + kk
    }
    for (int idx = tid; idx < 64 * 34; idx += 64) {
        const int c = idx / 34, r = idx % 34, rr = r / 17, t = r % 17;
        const int hh = h + rr, ww = j0 + t;
        float v = 0.0f;
        if (hh < HDIM && ww < WDIM)
            v = c2[((size_t)(n * 64 + c)) * HW + hh * WDIM + ww];
        const float a = g2[c] * istd2[c];
        sIn[c][rr][t] = fmaf(a, v, bt2[c] - mu2[c] * a);
    }
    __syncthreads();

    const int wave = tid >> 5, lane = tid & 31;
    const int m = lane & 15, hi = lane >> 4;
    const int o0 = wave << 4;

    v8f acc = {0.f, 0.f, 0.f, 0.f, 0.f, 0.f, 0.f, 0.f};
    #pragma unroll 4
    for (int c = 0; c < 64; ++c) {
        v2f a, b;
        a.x = sIn[c][hi][m];
        a.y = sIn[c][hi][m + 1];
        const int kk = (c << 2) + (hi << 1);
        b.x = sW[kk][o0 + m];
        b.y = sW[kk + 1][o0 + m];
        acc = __builtin_amdgcn_wmma_f32_16x16x4_f32(
            false, a, false, b, (short)0, acc, false, false);
    }
    const float bias = b3[o0 + m];
    float* dst = feat + ((size_t)(n * 32 + o0 + m)) * HW + h * WDIM + j0 + (hi << 3);
    #pragma unroll
    for (int r = 0; r < 8; ++r) dst[r] = acc[r] + bias;
}

// ---------------------------------------------------------------------------
// Pooled BN3 output: p[b, c*64 + ph*8 + pw] = avgpool16(BN3(feat)).
// One thread per pooled cell (16384 cells, 256 reads each).
// ---------------------------------------------------------------------------
__global__ __launch_bounds__(256) void pool_kernel(
    const float* __restrict__ feat, const float* __restrict__ mu3,
    const float* __restrict__ istd3, const float* __restrict__ g3,
    const float* __restrict__ bt3, float* __restrict__ p)
{
    const int cell = blockIdx.x * 256 + threadIdx.x;   // 0..16383
    const int b = cell >> 11, k = cell & 2047;
    const int c = k >> 6, ph = (k >> 3) & 7, pw = k & 7;
    const float* base = feat + ((size_t)(b * 32 + c)) * HW
                             + (ph * 16) * WDIM + pw * 16;
    float s = 0.0f;
    #pragma unroll 4
    for (int r = 0; r < 16; ++r)
        for (int t = 0; t < 16; ++t)
            s += base[r * WDIM + t];
    const float mean = s * (1.0f / 256.0f);
    const float a = g3[c] * istd3[c];
    p[cell] = fmaf(a, mean - mu3[c], bt3[c]);
}

// ---------------------------------------------------------------------------
// Head forward + analytic backward prep (single block):
//   z = p @ lw.T + lb ; r = relu(z) ; y = softmax(r)       -> d_out[0..15]
//   s[b,cls,m] = y[b,cls]*(d_{cls,m} - y[b,m]) * (z_m > 0)
//   gf[b,cls,k] = sum_m s*lw[m,k]
//   S1[b,cls,c] = g3_c * sum_{ph,pw} gf    (BN-backward sum, via pooling)
//   S2[b,cls,c] = sum_{ph,pw} gf * (p - bt3_c)   (g3 cancels against xhat)
// ---------------------------------------------------------------------------
__global__ __launch_bounds__(256) void head_kernel(
    const float* __restrict__ p, const float* __restrict__ lw,
    const float* __restrict__ lb, const float* __restrict__ g3,
    const float* __restrict__ bt3, float* __restrict__ y_out,
    float* __restrict__ gf, float* __restrict__ S1, float* __restrict__ S2)
{
    __shared__ float z[8][2];
    __shared__ float sc[8][2][2];
    const int tid = threadIdx.x;

    if (tid < 16) {
        const int b = tid >> 1, m2 = tid & 1;
        float acc = lb[m2];
        const float* pb = p + b * 2048;
        const float* w  = lw + m2 * 2048;
        for (int k = 0; k < 2048; ++k) acc = fmaf(pb[k], w[k], acc);
        z[b][m2] = acc;
    }
    __syncthreads();

    if (tid < 8) {
        const int b = tid;
        const float z0 = z[b][0], z1 = z[b][1];
        const float r0 = fmaxf(z0, 0.0f), r1 = fmaxf(z1, 0.0f);
        const float mx = fmaxf(r0, r1);
        const float e0 = __expf(r0 - mx), e1 = __expf(r1 - mx);
        const float inv = 1.0f / (e0 + e1);
        const float y0 = e0 * inv, y1 = e1 * inv;
        y_out[b * 2 + 0] = y0;
        y_out[b * 2 + 1] = y1;
        const float d0 = (z0 > 0.0f) ? 1.0f : 0.0f;
        const float d1 = (z1 > 0.0f) ? 1.0f : 0.0f;
        sc[b][0][0] = y0 * (1.0f - y0) * d0;  sc[b][0][1] = y0 * (-y1) * d1;
        sc[b][1][0] = y1 * (-y0) * d0;        sc[b][1][1] = y1 * (1.0f - y1) * d1;
    }
    __syncthreads();

    for (int idx = tid; idx < 8 * 2 * 2048; idx += 256) {
        const int b = idx >> 12, cls = (idx >> 11) & 1, k = idx & 2047;
        gf[idx] = sc[b][cls][0] * lw[k] + sc[b][cls][1] * lw[2048 + k];
    }
    __syncthreads();   // block-scope global visibility for gf

    for (int idx = tid; idx < 8 * 2 * 32; idx += 256) {
        const int b = idx >> 6, cls = (idx >> 5) & 1, c = idx & 31;
        const float* gfb = gf + (((size_t)(b * 2 + cls)) << 11) + (c << 6);
        const float* pb  = p + (((size_t)b) << 11) + (c << 6);
        float s1 = 0.0f, s2v = 0.0f;
        #pragma unroll 4
        for (int t = 0; t < 64; ++t) {
            const float g = gfb[t];
            s1  += g;
            s2v = fmaf(g, pb[t] - bt3[c], s2v);
        }
        S1[idx] = g3[c] * s1;
        S2[idx] = s2v;
    }
}

// ---------------------------------------------------------------------------
// CAM: cam[n,cls,h,w] = sum_c v*istd*(gx - S1/N) - v*(v-mu)*istd^2*S2/N,
// with gx = g3*gf/256 constant over each 16x16 pooled block.
// One block per (pooled cell, sample); 256 threads = the 16x16 pixels.
// ---------------------------------------------------------------------------
__global__ __launch_bounds__(256) void cam_kernel(
    const float* __restrict__ feat, const float* __restrict__ gf,
    const float* __restrict__ S1, const float* __restrict__ S2,
    const float* __restrict__ mu3, const float* __restrict__ istd3,
    const float* __restrict__ g3, float* __restrict__ cam)
{
    const int cellxy = blockIdx.x;               // 0..63 -> (ph, pw)
    const int ph = cellxy >> 3, pw = cellxy & 7;
    const int n = blockIdx.y;
    const int tid = threadIdx.x;
    const int ty = tid >> 4, tx = tid & 15;

    __shared__ float A[2][32], Bc[2][32], sMu[32];
    if (tid < 64) {
        const int cls = tid >> 5, c = tid & 31;
        const int k = (c << 6) + (ph << 3) + pw;
        const float gfv = gf[(((size_t)(n * 2 + cls)) << 11) + k];
        const float gx = g3[c] * gfv * (1.0f / 256.0f);
        const float is = istd3[c];
        const float invN = 1.0f / (float)(BATCH * HW);
        const float s1 = S1[(n * 2 + cls) * 32 + c];
        const float s2 = S2[(n * 2 + cls) * 32 + c];
        A[cls][c]  = is * (gx - s1 * invN);
        Bc[cls][c] = -is * is * s2 * invN;
    }
    if (tid < 32) sMu[tid] = mu3[tid];
    __syncthreads();

    const int h = (ph << 4) + ty, w = (pw << 4) + tx;
    float cam0 = 0.0f, cam1 = 0.0f;
    #pragma unroll 4
    for (int c = 0; c < 32; ++c) {
        const float v = feat[((size_t)(n * 32 + c)) * HW + h * WDIM + w];
        const float vvm = v * (v - sMu[c]);
        cam0 = fmaf(v, A[0][c], fmaf(vvm, Bc[0][c], cam0));
        cam1 = fmaf(v, A[1][c], fmaf(vvm, Bc[1][c], cam1));
    }
    const size_t o = (size_t)h * WDIM + w;
    cam[((size_t)(n * 2 + 0)) * HW + o] = cam0;
    cam[((size_t)(n * 2 + 1)) * HW + o] = cam1;
}

// ---------------------------------------------------------------------------
// Orchestration. Inputs in setup_inputs() order:
// 0:x 1:w1 2:b1 3:g1 4:bt1 5:w2 6:b2 7:g2 8:bt2 9:w3 10:b3 11:g3 12:bt3 13:lw 14:lb
// d_out: y (16 floats) then cams (8*2*128*128).
// ---------------------------------------------------------------------------
extern "C" void kernel_launch(void* const* d_in, const int* in_sizes, int n_in,
                              void* d_out, int out_size, void* d_ws, size_t ws_size,
                              hipStream_t stream) {
    const float* x   = (const float*)d_in[0];
    const float* w1  = (const float*)d_in[1];
    const float* b1  = (const float*)d_in[2];
    const float* g1  = (const float*)d_in[3];
    const float* bt1 = (const float*)d_in[4];
    const float* w2  = (const float*)d_in[5];
    const float* b2  = (const float*)d_in[6];
    const float* g2  = (const float*)d_in[7];
    const float* bt2 = (const float*)d_in[8];
    const float* w3  = (const float*)d_in[9];
    const float* b3  = (const float*)d_in[10];
    const float* g3  = (const float*)d_in[11];
    const float* bt3 = (const float*)d_in[12];
    const float* lw  = (const float*)d_in[13];
    const float* lb  = (const float*)d_in[14];

    float* ws    = (float*)d_ws;
    float* c1    = ws + OFF_C1;
    float* c2    = ws + OFF_C2;
    float* feat  = ws + OFF_FEAT;
    float* mu1   = ws + OFF_MU1;
    float* istd1 = ws + OFF_ISTD1;
    float* mu2   = ws + OFF_MU2;
    float* istd2 = ws + OFF_ISTD2;
    float* mu3   = ws + OFF_MU3;
    float* istd3 = ws + OFF_ISTD3;
    float* pbuf  = ws + OFF_P;
    float* gfbuf = ws + OFF_GF;
    float* S1b   = ws + OFF_S1;
    float* S2b   = ws + OFF_S2;

    float* yout  = (float*)d_out;
    float* cams  = (float*)d_out + 16;

    conv1_kernel<<<NPIX / 256, 256, 0, stream>>>(x, w1, b1, c1);
    bn_stats_kernel<<<16, 256, 0, stream>>>(c1, 16, mu1, istd1);
    conv2_wmma_kernel<<<NPIX / 16, 128, 0, stream>>>(c1, w2, b2, g1, bt1, mu1, istd1, c2);
    bn_stats_kernel<<<64, 256, 0, stream>>>(c2, 64, mu2, istd2);
    conv3_wmma_kernel<<<NPIX / 16, 64, 0, stream>>>(c2, w3, b3, g2, bt2, mu2, istd2, feat);
    bn_stats_kernel<<<32, 256, 0, stream>>>(feat, 32, mu3, istd3);
    pool_kernel<<<64, 256, 0, stream>>>(feat, mu3, istd3, g3, bt3, pbuf);
    head_kernel<<<1, 256, 0, stream>>>(pbuf, lw, lb, g3, bt3, yout, gfbuf, S1b, S2b);
    cam_kernel<<<dim3(64, 8), 256, 0, stream>>>(feat, gfbuf, S1b, S2b, mu3, istd3, g3, cams);
}